// GAT_75230647157566
// MI455X (gfx1250) — compile-verified
//
#include <hip/hip_runtime.h>
#include <math.h>

// ---------------- problem constants (from reference) ----------------
#define NNODES 50000
#define NEDGES 600000
#define EPLUS  (NEDGES + NNODES)   // edges + self loops
#define FIN    256
#define H1DIM  128                 // H * C
#define NH1    4
#define C1     32
#define OUTD   64

typedef float v2f __attribute__((ext_vector_type(2)));
typedef float v8f __attribute__((ext_vector_type(8)));

// ---------------- WMMA f32 GEMM: Hout[M,Ncols] = X[M,K] @ W[K,Ncols] ----------------
// One wave (32 threads) per block computes a 16x16 output tile with
// V_WMMA_F32_16X16X4_F32, looping K in steps of 4. The 16xK A-tile is a
// contiguous region of X (tile spans full K), staged in LDS with coalesced
// float4 loads.
__global__ __launch_bounds__(32)
void wmma_gemm_f32(const float* __restrict__ X, const float* __restrict__ W,
                   float* __restrict__ Hout, int K, int Ncols) {
    __shared__ float Atile[16 * 256];           // max K = 256 -> 16 KB
    const int lane = threadIdx.x;
    const int rowBase = blockIdx.x * 16;
    const int colBase = blockIdx.y * 16;

    // cooperative stage of contiguous 16*K floats
    const float4* src4 = reinterpret_cast<const float4*>(X + (size_t)rowBase * K);
    float4* dst4 = reinterpret_cast<float4*>(Atile);
    const int n4 = (16 * K) >> 2;
    for (int i = lane; i < n4; i += 32) dst4[i] = src4[i];
    __syncthreads();

    const int m    = lane & 15;
    const int half = lane >> 4;                 // 0: K pair {0,1}, 1: K pair {2,3}
    const int col  = colBase + m;

    v8f acc = {};
    for (int k0 = 0; k0 < K; k0 += 4) {
        const int ka = k0 + half * 2;
        v2f a;
        a.x = Atile[m * K + ka];
        a.y = Atile[m * K + ka + 1];
        v2f b;
        b.x = W[(size_t)ka * Ncols + col];
        b.y = W[(size_t)(ka + 1) * Ncols + col];
        acc = __builtin_amdgcn_wmma_f32_16x16x4_f32(
            /*neg_a=*/false, a, /*neg_b=*/false, b,
            /*c_mod=*/(short)0, acc, /*reuse_a=*/false, /*reuse_b=*/false);
    }
    // C/D layout: VGPR r holds row (half*8 + r), column = lane&15
    for (int r = 0; r < 8; ++r) {
        Hout[(size_t)(rowBase + half * 8 + r) * Ncols + col] = acc[r];
    }
}

// ---------------- per-node attention terms: als/ald[n,h] = sum_c h[n,h,c]*a[h,c] ----------------
__global__ void node_terms(const float* __restrict__ Hf, const float* __restrict__ a_s,
                           const float* __restrict__ a_d, float* __restrict__ als,
                           float* __restrict__ ald, int nNodes, int Hn, int Cc) {
    int idx = blockIdx.x * blockDim.x + threadIdx.x;    // idx = n*Hn + h
    if (idx >= nNodes * Hn) return;
    int h = idx % Hn;
    const float* hp = Hf + (size_t)idx * Cc;
    const float* as = a_s + h * Cc;
    const float* ad = a_d + h * Cc;
    float s0 = 0.f, s1 = 0.f;
    for (int c = 0; c < Cc; ++c) {
        float v = hp[c];
        s0 += v * as[c];
        s1 += v * ad[c];
    }
    als[idx] = s0;
    ald[idx] = s1;
}

// ---------------- fills ----------------
__global__ void fill_f32(float* __restrict__ p, float v, int n) {
    int i = blockIdx.x * blockDim.x + threadIdx.x;
    if (i < n) p[i] = v;
}
__global__ void fill_u32(unsigned* __restrict__ p, unsigned v, int n) {
    int i = blockIdx.x * blockDim.x + threadIdx.x;
    if (i < n) p[i] = v;
}

// ---------------- monotonic float<->uint encoding for atomicMax on floats ----------------
__device__ __forceinline__ unsigned ord_encode(float f) {
    unsigned u = __float_as_uint(f);
    return (u & 0x80000000u) ? ~u : (u | 0x80000000u);
}
__device__ __forceinline__ float ord_decode(unsigned e) {
    unsigned u = (e & 0x80000000u) ? (e ^ 0x80000000u) : ~e;
    return __uint_as_float(u);
}

// ---------------- edge pass 1: e = lrelu(als[src]+ald[dst]); segment max ----------------
__global__ void edge_scores_max(const int* __restrict__ srcI, const int* __restrict__ dstI,
                                const float* __restrict__ als, const float* __restrict__ ald,
                                float* __restrict__ ebuf, unsigned* __restrict__ mEnc,
                                int nE, int nNodes, int Hn) {
    int e = blockIdx.x * blockDim.x + threadIdx.x;
    if (e >= nE + nNodes) return;
    int s = (e < nE) ? srcI[e] : (e - nE);
    int d = (e < nE) ? dstI[e] : (e - nE);
    for (int h = 0; h < Hn; ++h) {
        float v = als[s * Hn + h] + ald[d * Hn + h];
        v = (v > 0.f) ? v : 0.2f * v;                   // leaky_relu 0.2
        ebuf[(size_t)e * Hn + h] = v;
        atomicMax(&mEnc[d * Hn + h], ord_encode(v));
    }
}

// ---------------- edge pass 2: p = exp(e - m[dst]); segment sum ----------------
__global__ void edge_exp_sum(const int* __restrict__ srcI, const int* __restrict__ dstI,
                             float* __restrict__ ebuf, const unsigned* __restrict__ mEnc,
                             float* __restrict__ sbuf, int nE, int nNodes, int Hn) {
    int e = blockIdx.x * blockDim.x + threadIdx.x;
    if (e >= nE + nNodes) return;
    int d = (e < nE) ? dstI[e] : (e - nE);
    for (int h = 0; h < Hn; ++h) {
        float m = ord_decode(mEnc[d * Hn + h]);
        float p = __expf(ebuf[(size_t)e * Hn + h] - m);
        ebuf[(size_t)e * Hn + h] = p;
        atomicAdd(&sbuf[d * Hn + h], p);
    }
}

// ---------------- edge pass 3: agg[dst,h,:] += (p/s[dst,h]) * h[src,h,:] ----------------
__global__ void edge_aggregate(const int* __restrict__ srcI, const int* __restrict__ dstI,
                               const float* __restrict__ ebuf, const float* __restrict__ sbuf,
                               const float* __restrict__ Hf, float* __restrict__ agg,
                               int nE, int nNodes, int Hn, int Cc) {
    int idx = blockIdx.x * blockDim.x + threadIdx.x;    // idx = e*Hn + h
    int total = (nE + nNodes) * Hn;
    if (idx >= total) return;
    int e = idx / Hn, h = idx % Hn;
    int s = (e < nE) ? srcI[e] : (e - nE);
    int d = (e < nE) ? dstI[e] : (e - nE);
    float alpha = ebuf[idx] / sbuf[d * Hn + h];
    const float* hp = Hf  + ((size_t)s * Hn + h) * Cc;
    float*       op = agg + ((size_t)d * Hn + h) * Cc;
    for (int c = 0; c < Cc; ++c) atomicAdd(&op[c], alpha * hp[c]);
}

// ---------------- elementwise: out = lrelu(agg + b) / out = agg + b ----------------
__global__ void bias_lrelu(const float* __restrict__ agg, const float* __restrict__ b,
                           float* __restrict__ out, int n, int Dim) {
    int i = blockIdx.x * blockDim.x + threadIdx.x;
    if (i >= n) return;
    float v = agg[i] + b[i % Dim];
    out[i] = (v > 0.f) ? v : 0.2f * v;
}
__global__ void bias_add(const float* __restrict__ agg, const float* __restrict__ b,
                         float* __restrict__ out, int n, int Dim) {
    int i = blockIdx.x * blockDim.x + threadIdx.x;
    if (i >= n) return;
    out[i] = agg[i] + b[i % Dim];
}

// ---------------- launch ----------------
extern "C" void kernel_launch(void* const* d_in, const int* in_sizes, int n_in,
                              void* d_out, int out_size, void* d_ws, size_t ws_size,
                              hipStream_t stream) {
    const float* x    = (const float*)d_in[0];
    const int*   ei   = (const int*)  d_in[1];
    const float* W1   = (const float*)d_in[2];
    const float* a_s1 = (const float*)d_in[3];
    const float* a_d1 = (const float*)d_in[4];
    const float* b1   = (const float*)d_in[5];
    const float* W2   = (const float*)d_in[6];
    const float* a_s2 = (const float*)d_in[7];
    const float* a_d2 = (const float*)d_in[8];
    const float* b2   = (const float*)d_in[9];
    float* out = (float*)d_out;

    const int* srcI = ei;
    const int* dstI = ei + NEDGES;

    // workspace layout (floats)
    float* ws   = (float*)d_ws;
    float* h1   = ws;                                   // N*128
    float* agg  = h1   + (size_t)NNODES * H1DIM;        // N*128 (layer1 agg; later h2 + agg2)
    float* ebuf = agg  + (size_t)NNODES * H1DIM;        // EPLUS*4
    float* als  = ebuf + (size_t)EPLUS * NH1;           // N*4
    float* ald  = als  + (size_t)NNODES * NH1;          // N*4
    unsigned* mEnc = (unsigned*)(ald + (size_t)NNODES * NH1);  // N*4 (uint view)
    float* sbuf = (float*)mEnc + (size_t)NNODES * NH1;  // N*4

    const int T = 256;
    auto blocks = [](long n, int t) { return (int)((n + t - 1) / t); };

    // ================= layer 1 (H=4, C=32) =================
    {
        dim3 g(NNODES / 16, H1DIM / 16);
        wmma_gemm_f32<<<g, 32, 0, stream>>>(x, W1, h1, FIN, H1DIM);
    }
    node_terms<<<blocks((long)NNODES * NH1, T), T, 0, stream>>>(h1, a_s1, a_d1, als, ald, NNODES, NH1, C1);
    fill_u32<<<blocks((long)NNODES * NH1, T), T, 0, stream>>>(mEnc, 0u, NNODES * NH1);
    fill_f32<<<blocks((long)NNODES * NH1, T), T, 0, stream>>>(sbuf, 0.f, NNODES * NH1);
    fill_f32<<<blocks((long)NNODES * H1DIM, T), T, 0, stream>>>(agg, 0.f, NNODES * H1DIM);
    edge_scores_max<<<blocks((long)EPLUS, T), T, 0, stream>>>(srcI, dstI, als, ald, ebuf, mEnc, NEDGES, NNODES, NH1);
    edge_exp_sum<<<blocks((long)EPLUS, T), T, 0, stream>>>(srcI, dstI, ebuf, mEnc, sbuf, NEDGES, NNODES, NH1);
    edge_aggregate<<<blocks((long)EPLUS * NH1, T), T, 0, stream>>>(srcI, dstI, ebuf, sbuf, h1, agg, NEDGES, NNODES, NH1, C1);
    // x2 = lrelu(agg + b1)  -> reuse h1 buffer
    bias_lrelu<<<blocks((long)NNODES * H1DIM, T), T, 0, stream>>>(agg, b1, h1, NNODES * H1DIM, H1DIM);

    // ================= layer 2 (H=1, C=64, concat=False -> mean over 1 head = identity) =================
    float* h2   = agg;                                  // N*64
    float* agg2 = agg + (size_t)NNODES * OUTD;          // N*64 (second half of agg region)
    {
        dim3 g(NNODES / 16, OUTD / 16);
        wmma_gemm_f32<<<g, 32, 0, stream>>>(h1, W2, h2, H1DIM, OUTD);
    }
    node_terms<<<blocks((long)NNODES, T), T, 0, stream>>>(h2, a_s2, a_d2, als, ald, NNODES, 1, OUTD);
    fill_u32<<<blocks((long)NNODES, T), T, 0, stream>>>(mEnc, 0u, NNODES);
    fill_f32<<<blocks((long)NNODES, T), T, 0, stream>>>(sbuf, 0.f, NNODES);
    fill_f32<<<blocks((long)NNODES * OUTD, T), T, 0, stream>>>(agg2, 0.f, NNODES * OUTD);
    edge_scores_max<<<blocks((long)EPLUS, T), T, 0, stream>>>(srcI, dstI, als, ald, ebuf, mEnc, NEDGES, NNODES, 1);
    edge_exp_sum<<<blocks((long)EPLUS, T), T, 0, stream>>>(srcI, dstI, ebuf, mEnc, sbuf, NEDGES, NNODES, 1);
    edge_aggregate<<<blocks((long)EPLUS, T), T, 0, stream>>>(srcI, dstI, ebuf, sbuf, h2, agg2, NEDGES, NNODES, 1, OUTD);
    bias_add<<<blocks((long)NNODES * OUTD, T), T, 0, stream>>>(agg2, b2, out, NNODES * OUTD, OUTD);
}